// QueryAndGroup_27608049778803
// MI455X (gfx1250) — compile-verified
//
#include <hip/hip_runtime.h>
#include <stdint.h>

#define BDIM    256
#define WAVES   8
#define WTILE   16      // centroids per wave (WMMA M)
#define PTILE   16      // points per WMMA (N)
#define NPTS    16384   // N (whole batch staged in LDS)
#define RANGES  4       // N-scan split across waves
#define CTILES  2       // centroid tiles per block
#define NSAMPLE 32
#define RADIUS2 0.01f

typedef float v2f __attribute__((ext_vector_type(2)));
typedef float v8f __attribute__((ext_vector_type(8)));
typedef int   v4i __attribute__((vector_size(16)));
typedef __attribute__((address_space(1))) v4i* gv4i_ptr;   // global int4*
typedef __attribute__((address_space(3))) v4i* lv4i_ptr;   // LDS int4*

static __device__ __forceinline__ unsigned ballot32(bool p) {
#if __has_builtin(__builtin_amdgcn_ballot_w32)
  return __builtin_amdgcn_ballot_w32(p);
#else
  return (unsigned)__ballot(p);
#endif
}

// Async global -> LDS copy of 16 bytes (tracked by ASYNCcnt).
static __device__ __forceinline__ void async_copy_b128(const float* gsrc, float* ldst) {
#if __has_builtin(__builtin_amdgcn_global_load_async_to_lds_b128)
  __builtin_amdgcn_global_load_async_to_lds_b128(
      (gv4i_ptr)gsrc, (lv4i_ptr)ldst, 0, 0);
#else
  unsigned loff = (unsigned)(uintptr_t)ldst;   // low 32 bits = LDS byte offset
  asm volatile("global_load_async_to_lds_b128 %0, %1, off"
               :: "v"(loff), "v"(gsrc) : "memory");
#endif
}

static __device__ __forceinline__ void wait_async0() {
#if __has_builtin(__builtin_amdgcn_s_wait_asynccnt)
  __builtin_amdgcn_s_wait_asynccnt(0);
#else
  asm volatile("s_wait_asynccnt 0" ::: "memory");
#endif
}

__global__ __launch_bounds__(BDIM) void ballquery_group_wmma(
    const float* __restrict__ points,     // (B,N,3)
    const float* __restrict__ centroids,  // (B,K,3)
    const float* __restrict__ features,   // (B,C,N)
    float* __restrict__ outXyz,           // (B,3,K,S)
    float* __restrict__ outFeat,          // (B,C,K,S)
    int Nn, int Kn, int Cn)
{
  __shared__ __align__(16) float ldsPts[NPTS * 3];            // 192 KB: whole batch
  __shared__ int idxBuf[WAVES][WTILE * NSAMPLE];              // 16 KB: per (range,tile)
  __shared__ int cntBuf[WAVES][WTILE];                        // 512 B

  const int tid  = threadIdx.x;
  const int wave = tid >> 5;
  const int lane = tid & 31;
  const int m    = lane & 15;

  const int blocksPerB = Kn / (CTILES * WTILE);               // 64
  const int b    = blockIdx.x / blocksPerB;
  const int blkK = (blockIdx.x % blocksPerB) * (CTILES * WTILE);

  const int t = wave & (CTILES - 1);                          // centroid tile 0/1
  const int r = wave >> 1;                                    // scan range 0..3
  const int kBase = blkK + t * WTILE;

  // ---- stage ALL N points of this batch into LDS via one async DMA burst ----
  {
    const float* gsrc = points + (size_t)b * Nn * 3;
    const int nB128 = (NPTS * 3) / 4;                         // 12288 transfers
#pragma unroll
    for (int j0 = 0; j0 < nB128; j0 += BDIM) {                // 48 per thread (<63 ASYNCcnt)
      int j = j0 + tid;
      async_copy_b128(gsrc + j * 4, &ldsPts[j * 4]);
    }
    wait_async0();
    __syncthreads();
  }

  // A fragment (16x4 f32), row m = [cx, cy, cz, |c|^2]:
  //   lanes 0-15 hold K=0,1 ; lanes 16-31 hold K=2,3
  const float* cptr = centroids + ((size_t)b * Kn + kBase + m) * 3;
  const float cx = cptr[0], cy = cptr[1], cz = cptr[2];
  const float c2 = cx * cx + cy * cy + cz * cz;
  v2f afrag;
  afrag[0] = (lane < 16) ? cx : cz;
  afrag[1] = (lane < 16) ? cy : c2;

  // ---- WMMA distance scan over this wave's contiguous quarter of N ----
  int  cnt   = (lane < 16) ? 0 : NSAMPLE;   // upper half: permanently "done"
  int* myIdx = &idxBuf[wave][0];
  const int rBase = r * (NPTS / RANGES);                      // 4096-point range

  for (int tile = 0; tile < NPTS / RANGES; tile += PTILE) {
    if (ballot32(cnt >= NSAMPLE) == 0xFFFFFFFFu) break;       // wave-uniform exit
    const float* p = &ldsPts[(size_t)(rBase + tile + m) * 3];
    const float px = p[0], py = p[1], pz = p[2];
    const float p2 = px * px + py * py + pz * pz;
    // B fragment (4x16 f32), col n = [-2px, -2py, -2pz, 1]
    v2f bfrag;
    bfrag[0] = (lane < 16) ? -2.0f * px : -2.0f * pz;
    bfrag[1] = (lane < 16) ? -2.0f * py : 1.0f;
    v8f cz8 = {};
    v8f d = __builtin_amdgcn_wmma_f32_16x16x4_f32(
        false, afrag, false, bfrag, (short)0, cz8, false, false);

    // D VGPR j: lanes 0-15 = row j, lanes 16-31 = row j+8 (cols = lane%16).
    unsigned mym = 0;
#pragma unroll
    for (int j = 0; j < 8; ++j) {
      unsigned bj = ballot32(d[j] + p2 < RADIUS2);
      mym = (lane == j)     ? (bj & 0xFFFFu) : mym;
      mym = (lane == j + 8) ? (bj >> 16)     : mym;
    }
    const int nBase = rBase + tile;
    while (mym && cnt < NSAMPLE) {                            // in-index-order append
      int c = __builtin_ctz(mym);
      mym &= mym - 1;
      myIdx[lane * NSAMPLE + cnt] = nBase + c;
      ++cnt;
    }
  }
  if (lane < 16) cntBuf[wave][lane] = cnt;
  __syncthreads();

  // ---- merge 4 ordered range lists + gather (each wave: 4 centroids) ----
  const size_t KS = (size_t)Kn * NSAMPLE;
  for (int q = 0; q < (CTILES * WTILE) / WAVES; ++q) {
    const int cIdx = wave * 4 + q;                            // 0..31 within block
    const int tt = cIdx >> 4, ii = cIdx & 15;
    const int k = blkK + cIdx;
    const int* b0 = &idxBuf[0 * CTILES + tt][ii * NSAMPLE];
    const int* b1 = &idxBuf[1 * CTILES + tt][ii * NSAMPLE];
    const int* b2 = &idxBuf[2 * CTILES + tt][ii * NSAMPLE];
    const int* b3 = &idxBuf[3 * CTILES + tt][ii * NSAMPLE];
    const int c0 = cntBuf[0 * CTILES + tt][ii];
    const int c1 = cntBuf[1 * CTILES + tt][ii];
    const int c2i = cntBuf[2 * CTILES + tt][ii];
    const int c3 = cntBuf[3 * CTILES + tt][ii];
    const int o2 = c0 + c1, o3 = o2 + c2i, tot = o3 + c3;
    const int s = lane;
    int idx;
    if      (s < c0)  idx = b0[s];
    else if (s < o2)  idx = b1[s - c0];
    else if (s < o3)  idx = b2[s - o2];
    else if (s < tot) idx = b3[s - o3];
    else {            // pad with first overall hit, or 0 if none
      idx = (c0 > 0) ? b0[0] : (c1 > 0) ? b1[0] : (c2i > 0) ? b2[0]
          : (c3 > 0) ? b3[0] : 0;
    }

    const float* pp = points    + ((size_t)b * Nn + idx) * 3;
    const float* cc = centroids + ((size_t)b * Kn + k) * 3;
    const size_t xo = ((size_t)b * 3 * Kn + k) * NSAMPLE + lane;
    outXyz[xo]          = pp[0] - cc[0];
    outXyz[xo + KS]     = pp[1] - cc[1];
    outXyz[xo + 2 * KS] = pp[2] - cc[2];
    const float* fsrc = features + (size_t)b * Cn * Nn + idx;
    float*       fdst = outFeat + ((size_t)b * Cn * Kn + k) * NSAMPLE + lane;
#pragma unroll 4
    for (int c = 0; c < Cn; ++c)
      fdst[(size_t)c * KS] = fsrc[(size_t)c * Nn];
  }
}

extern "C" void kernel_launch(void* const* d_in, const int* in_sizes, int n_in,
                              void* d_out, int out_size, void* d_ws, size_t ws_size,
                              hipStream_t stream) {
  (void)in_sizes; (void)n_in; (void)out_size; (void)d_ws; (void)ws_size;
  const float* points    = (const float*)d_in[0];
  const float* centroids = (const float*)d_in[1];
  const float* features  = (const float*)d_in[2];
  const int B = 4, N = 16384, K = 2048, C = 64;
  float* outXyz  = (float*)d_out;
  float* outFeat = outXyz + (size_t)B * 3 * K * NSAMPLE;
  const int blocks = B * (K / (CTILES * WTILE));              // 256 blocks x 8 waves
  ballquery_group_wmma<<<blocks, BDIM, 0, stream>>>(
      points, centroids, features, outXyz, outFeat, N, K, C);
}